// Classifier_Net_45569603010819
// MI455X (gfx1250) — compile-verified
//
#include <hip/hip_runtime.h>
#include <hip/hip_bf16.h>

// ---------------------------------------------------------------------------
// Sizes from the reference
// ---------------------------------------------------------------------------
#define BATCH 8192
#define FL    96
#define HALFW 288          // 3*FL
#define ROWW  576          // 6*FL
#define NROWS (2*BATCH)    // (b, half) fusion tasks = 16384
#define GDIM  192          // fusion feature width (2*FL)
#define CLK   576          // cl GEMM K
#define NCH   288          // resh/cl output channels
#define EPS   1e-5f

typedef float v2f __attribute__((ext_vector_type(2)));
typedef float v8f __attribute__((ext_vector_type(8)));
typedef unsigned int v4u __attribute__((ext_vector_type(4)));
typedef unsigned int v8u __attribute__((ext_vector_type(8)));

__device__ __forceinline__ float lrelu(float v) { return v >= 0.f ? v : 0.1f * v; }

// ===========================================================================
// Kernel 1: per-(b,half) fusion.  96 threads = 3 waves (wave32).
//   - half-row staged into LDS by the Tensor Data Mover (1-D tile descriptor,
//     issued once by wave 0; TENSORcnt waited before use)
//   - conv chains (2 layers x 3 segments) in LDS
//   - dynamic 3x3 kernel from bert via de_W (96 -> 9)
//   - outer product + lrelu + 3x3 stencil + lrelu, row/col means
//   Column sums use per-wave LDS banks with a skewed j schedule: conflict-free
//   and deterministic (fixed accumulation order), no atomics needed.
// ===========================================================================
__global__ __launch_bounds__(96) void fusion_kernel(
    const float* __restrict__ x,
    const float* __restrict__ cw1, const float* __restrict__ cb1,
    const float* __restrict__ cw2, const float* __restrict__ cb2,
    const float* __restrict__ cw3, const float* __restrict__ cb3,
    const float* __restrict__ cw4, const float* __restrict__ cb4,
    const float* __restrict__ cw5, const float* __restrict__ cb5,
    const float* __restrict__ cw6, const float* __restrict__ cb6,
    const float* __restrict__ de_W, const float* __restrict__ de_b,
    float* __restrict__ g_out)
{
    __shared__ float s_in[HALFW];        // raw half-row (TDM destination)
    __shared__ float s_t [HALFW];        // after conv layer 1
    __shared__ float s_u [HALFW];        // arm | x86 | bert
    __shared__ float s_kk[9];            // dynamic 3x3 kernel
    __shared__ float s_colw[3][FL];      // per-wave column accumulators

    const int i    = threadIdx.x;        // 0..95
    const int r    = blockIdx.x;         // row id = 2*b + half
    const int b    = r >> 1;
    const int half = r & 1;
    const float* xrow = x + (size_t)b * ROWW + half * HALFW;

    // zero the 3*96 = 288 column-accumulator slots
    for (int idx = i; idx < HALFW; idx += 96)
        (&s_colw[0][0])[idx] = 0.f;

    // ---- TDM: async 1-D tile load of 288 contiguous f32 into LDS ----------
    // D# group0: [1:0]=count=1 | [63:32]=lds_addr | [120:64]=global_addr |
    //            [127:126]=type=2.  group1: data_size=4B, tensor_dim0 =
    //            tile_dim0 = 288, 1-D tile (tile_dim1 = 0), no pad/iterate.
    if (i < 32) {                        // wave 0 only (TDM ignores EXEC)
        unsigned lds_off = (unsigned)(size_t)(void*)s_in;     // LDS byte offset
        unsigned long long ga = (unsigned long long)(const void*)xrow;
        v4u g0;
        g0.x = 1u;                                            // count = 1
        g0.y = lds_off;                                       // lds_addr
        g0.z = (unsigned)ga;                                  // global_addr lo
        g0.w = (unsigned)((ga >> 32) & 0x01FFFFFFull) | (2u << 30); // hi | type=2
        v8u g1;
        g1[0] = (2u << 16);               // data_size = 2 (4 bytes)
        g1[1] = ((unsigned)HALFW) << 16;  // tensor_dim0 = 288 (lo16)
        g1[2] = (1u << 16);               // dim0 hi16 = 0 | tensor_dim1 = 1
        g1[3] = ((unsigned)HALFW) << 16;  // dim1 hi16 = 0 | tile_dim0 = 288
        g1[4] = 0u;                       // tile_dim1 = tile_dim2 = 0
        g1[5] = (unsigned)HALFW;          // tensor_dim0_stride = 288 (lo32)
        g1[6] = 0u;
        g1[7] = 0u;
        asm volatile("tensor_load_to_lds %0, %1" :: "s"(g0), "s"(g1) : "memory");
        __builtin_amdgcn_s_wait_tensorcnt(0);
    }
    __syncthreads();

    // ---- conv layer 1 (cw1/2/3) : s_in -> s_t ----
    {
        const float* cw[3] = {cw1, cw2, cw3};
        const float* cb[3] = {cb1, cb2, cb3};
#pragma unroll
        for (int s = 0; s < 3; ++s) {
            const float w0 = cw[s][0], w1 = cw[s][1], w2 = cw[s][2], bb = cb[s][0];
            const float* seg = s_in + s * FL;
            float l = (i > 0)  ? seg[i - 1] : 0.f;
            float m = seg[i];
            float rr = (i < FL - 1) ? seg[i + 1] : 0.f;
            s_t[s * FL + i] = lrelu(w0 * l + w1 * m + w2 * rr + bb);
        }
    }
    __syncthreads();

    // ---- conv layer 2 (cw4/5/6) : s_t -> s_u  (arm | x86 | bert) ----
    {
        const float* cw[3] = {cw4, cw5, cw6};
        const float* cb[3] = {cb4, cb5, cb6};
#pragma unroll
        for (int s = 0; s < 3; ++s) {
            const float w0 = cw[s][0], w1 = cw[s][1], w2 = cw[s][2], bb = cb[s][0];
            const float* seg = s_t + s * FL;
            float l = (i > 0)  ? seg[i - 1] : 0.f;
            float m = seg[i];
            float rr = (i < FL - 1) ? seg[i + 1] : 0.f;
            s_u[s * FL + i] = lrelu(w0 * l + w1 * m + w2 * rr + bb);
        }
    }
    __syncthreads();

    // ---- dynamic 3x3 kernel: kk[p] = lrelu(bert . de_W[p] + de_b[p]) ----
    if (i < 9) {
        float acc = de_b[i];
        const float* wrow = de_W + i * FL;
        const float* bert = s_u + 2 * FL;
#pragma unroll 8
        for (int l = 0; l < FL; ++l) acc += bert[l] * wrow[l];
        s_kk[i] = lrelu(acc);
    }
    __syncthreads();

    const float k00 = s_kk[0], k01 = s_kk[1], k02 = s_kk[2];
    const float k10 = s_kk[3], k11 = s_kk[4], k12 = s_kk[5];
    const float k20 = s_kk[6], k21 = s_kk[7], k22 = s_kk[8];

    const float am = (i > 0)      ? s_u[i - 1] : 0.f;  // arm[i-1]
    const float a0 = s_u[i];                            // arm[i]
    const float ap = (i < FL - 1) ? s_u[i + 1] : 0.f;  // arm[i+1]
    const float* bx = s_u + FL;                         // x86
    float* colbank = s_colw[i >> 5];                    // wave-private bank

    float rowsum = 0.f;
    // skewed schedule: within a wave, lanes hit 32 consecutive j mod 96 ->
    // distinct addresses, no bank conflicts, no races, deterministic order.
    for (int t = 0; t < FL; ++t) {
        int j = i + t; if (j >= FL) j -= FL;
        float bm = (j > 0)      ? bx[j - 1] : 0.f;
        float b0 = bx[j];
        float bp = (j < FL - 1) ? bx[j + 1] : 0.f;
        float fe = k00 * lrelu(am * bm) + k01 * lrelu(am * b0) + k02 * lrelu(am * bp)
                 + k10 * lrelu(a0 * bm) + k11 * lrelu(a0 * b0) + k12 * lrelu(a0 * bp)
                 + k20 * lrelu(ap * bm) + k21 * lrelu(ap * b0) + k22 * lrelu(ap * bp);
        fe = lrelu(fe);
        rowsum += fe;
        colbank[j] += fe;
    }
    __syncthreads();

    float* grow = g_out + (size_t)r * GDIM;
    grow[i]      = rowsum * (1.f / FL);
    grow[FL + i] = (s_colw[0][i] + s_colw[1][i] + s_colw[2][i]) * (1.f / FL);
}

// ===========================================================================
// fp32 WMMA fragment layouts (CDNA5 ISA 7.12.2, wave32):
//   A 16x4 : lane l, vgpr v  -> m = l%16, k = 2*(l/16)+v
//   B 4x16 : lane l, vgpr v  -> n = l%16, k = 2*(l/16)+v   (B[k][n])
//   C 16x16: lane l, vgpr v  -> m = v + 8*(l/16), n = l%16
// Each wave computes a 16x32 strip (two n-tiles) to reuse the A fragment.
// ===========================================================================

// Kernel 2: resh GEMM.  e[r] = lrelu(BN(g[r] @ rs_W^T + rs_b)) + x-resid
//   M = 16384 (r = 2b+half), K = 192, N = 288.
__global__ __launch_bounds__(256) void resh_gemm_kernel(
    const float* __restrict__ g,    const float* __restrict__ rs_W,
    const float* __restrict__ rs_b, const float* __restrict__ rs_g,
    const float* __restrict__ rs_be,const float* __restrict__ rs_m,
    const float* __restrict__ rs_v, const float* __restrict__ x,
    float* __restrict__ ce)
{
    const int NTP  = NCH / 32;                               // 9 n-tile pairs
    const int wave = (blockIdx.x * blockDim.x + threadIdx.x) >> 5;
    if (wave >= (NROWS / 16) * NTP) return;
    const int mt = wave / NTP, ntp = wave % NTP;
    const int lane    = threadIdx.x & 31;
    const int lane_lo = lane & 15;
    const int lane_hi = lane >> 4;

    const float* apA  = g    + (size_t)(mt * 16 + lane_lo) * GDIM + 2 * lane_hi;
    const float* apB0 = rs_W + (size_t)(ntp * 32 + lane_lo) * GDIM + 2 * lane_hi;
    const float* apB1 = apB0 + 16 * GDIM;

    v8f acc0 = {}, acc1 = {};
#pragma unroll 8
    for (int k0 = 0; k0 < GDIM; k0 += 4) {
        v2f a  = *(const v2f*)(apA  + k0);
        v2f b0 = *(const v2f*)(apB0 + k0);
        v2f b1 = *(const v2f*)(apB1 + k0);
        acc0 = __builtin_amdgcn_wmma_f32_16x16x4_f32(
                   false, a, false, b0, (short)0, acc0, false, false);
        acc1 = __builtin_amdgcn_wmma_f32_16x16x4_f32(
                   false, a, false, b1, (short)0, acc1, false, false);
    }

#pragma unroll
    for (int tt = 0; tt < 2; ++tt) {
        v8f acc = tt ? acc1 : acc0;
        const int   n     = ntp * 32 + tt * 16 + lane_lo;
        const float bias  = rs_b[n];
        const float scale = rs_g[n] * rsqrtf(rs_v[n] + EPS);
        const float shift = rs_be[n] - rs_m[n] * scale;
#pragma unroll
        for (int v = 0; v < 8; ++v) {
            int r  = mt * 16 + v + 8 * lane_hi;
            int b  = r >> 1, half = r & 1;
            size_t o = (size_t)b * ROWW + half * HALFW + n;
            float val = (acc[v] + bias) * scale + shift;
            ce[o] = lrelu(val) + x[o];
        }
    }
}

// Kernel 3: classifier GEMM.  h = relu(BN(ce @ cl_W1^T + cl_b1))
//   M = 8192, K = 576, N = 288.
__global__ __launch_bounds__(256) void cl_gemm_kernel(
    const float* __restrict__ ce,    const float* __restrict__ cl_W1,
    const float* __restrict__ cl_b1, const float* __restrict__ cl_g,
    const float* __restrict__ cl_be, const float* __restrict__ cl_m,
    const float* __restrict__ cl_v,  float* __restrict__ h)
{
    const int NTP  = NCH / 32;                               // 9
    const int wave = (blockIdx.x * blockDim.x + threadIdx.x) >> 5;
    if (wave >= (BATCH / 16) * NTP) return;
    const int mt = wave / NTP, ntp = wave % NTP;
    const int lane    = threadIdx.x & 31;
    const int lane_lo = lane & 15;
    const int lane_hi = lane >> 4;

    const float* apA  = ce    + (size_t)(mt * 16 + lane_lo) * CLK + 2 * lane_hi;
    const float* apB0 = cl_W1 + (size_t)(ntp * 32 + lane_lo) * CLK + 2 * lane_hi;
    const float* apB1 = apB0 + 16 * CLK;

    v8f acc0 = {}, acc1 = {};
#pragma unroll 8
    for (int k0 = 0; k0 < CLK; k0 += 4) {
        v2f a  = *(const v2f*)(apA  + k0);
        v2f b0 = *(const v2f*)(apB0 + k0);
        v2f b1 = *(const v2f*)(apB1 + k0);
        acc0 = __builtin_amdgcn_wmma_f32_16x16x4_f32(
                   false, a, false, b0, (short)0, acc0, false, false);
        acc1 = __builtin_amdgcn_wmma_f32_16x16x4_f32(
                   false, a, false, b1, (short)0, acc1, false, false);
    }

#pragma unroll
    for (int tt = 0; tt < 2; ++tt) {
        v8f acc = tt ? acc1 : acc0;
        const int   n     = ntp * 32 + tt * 16 + lane_lo;
        const float bias  = cl_b1[n];
        const float scale = cl_g[n] * rsqrtf(cl_v[n] + EPS);
        const float shift = cl_be[n] - cl_m[n] * scale;
#pragma unroll
        for (int v = 0; v < 8; ++v) {
            int r = mt * 16 + v + 8 * lane_hi;
            float val = (acc[v] + bias) * scale + shift;
            h[(size_t)r * NCH + n] = fmaxf(val, 0.f);       // relu
        }
    }
}

// Kernel 4: 288 -> 2 head.  One thread per batch row.
__global__ __launch_bounds__(256) void head_kernel(
    const float* __restrict__ h, const float* __restrict__ W2,
    const float* __restrict__ b2, float* __restrict__ out)
{
    int b = blockIdx.x * blockDim.x + threadIdx.x;
    if (b >= BATCH) return;
    const float* hr = h + (size_t)b * NCH;
    float a0 = 0.f, a1 = 0.f;
#pragma unroll 4
    for (int c = 0; c < NCH; ++c) {
        float hv = hr[c];
        a0 += hv * W2[c];
        a1 += hv * W2[NCH + c];
    }
    out[2 * b + 0] = a0 + b2[0];
    out[2 * b + 1] = a1 + b2[1];
}

// ===========================================================================
extern "C" void kernel_launch(void* const* d_in, const int* in_sizes, int n_in,
                              void* d_out, int out_size, void* d_ws, size_t ws_size,
                              hipStream_t stream) {
    const float* x    = (const float*)d_in[0];
    const float* cw1  = (const float*)d_in[1];  const float* cb1 = (const float*)d_in[2];
    const float* cw2  = (const float*)d_in[3];  const float* cb2 = (const float*)d_in[4];
    const float* cw3  = (const float*)d_in[5];  const float* cb3 = (const float*)d_in[6];
    const float* cw4  = (const float*)d_in[7];  const float* cb4 = (const float*)d_in[8];
    const float* cw5  = (const float*)d_in[9];  const float* cb5 = (const float*)d_in[10];
    const float* cw6  = (const float*)d_in[11]; const float* cb6 = (const float*)d_in[12];
    const float* de_W = (const float*)d_in[13]; const float* de_b = (const float*)d_in[14];
    const float* rs_W = (const float*)d_in[15]; const float* rs_b = (const float*)d_in[16];
    const float* rs_g = (const float*)d_in[17]; const float* rs_be = (const float*)d_in[18];
    const float* rs_m = (const float*)d_in[19]; const float* rs_v = (const float*)d_in[20];
    const float* cl_W1 = (const float*)d_in[21]; const float* cl_b1 = (const float*)d_in[22];
    const float* cl_g  = (const float*)d_in[23]; const float* cl_be = (const float*)d_in[24];
    const float* cl_m  = (const float*)d_in[25]; const float* cl_v  = (const float*)d_in[26];
    const float* cl_W2 = (const float*)d_in[27]; const float* cl_b2 = (const float*)d_in[28];

    // workspace: g (16384x192) | ce (8192x576) | h (8192x288)  ~39 MB
    float* g  = (float*)d_ws;
    float* ce = g  + (size_t)NROWS * GDIM;
    float* h  = ce + (size_t)BATCH * ROWW;

    fusion_kernel<<<NROWS, 96, 0, stream>>>(
        x, cw1, cb1, cw2, cb2, cw3, cb3, cw4, cb4, cw5, cb5, cw6, cb6,
        de_W, de_b, g);

    {   // 1024 m-tiles x 9 n-tile-pairs = 9216 waves, 8 waves/block
        int waves = (NROWS / 16) * (NCH / 32);
        resh_gemm_kernel<<<(waves + 7) / 8, 256, 0, stream>>>(
            g, rs_W, rs_b, rs_g, rs_be, rs_m, rs_v, x, ce);
    }
    {   // 512 m-tiles x 9 n-tile-pairs = 4608 waves
        int waves = (BATCH / 16) * (NCH / 32);
        cl_gemm_kernel<<<(waves + 7) / 8, 256, 0, stream>>>(
            ce, cl_W1, cl_b1, cl_g, cl_be, cl_m, cl_v, h);
    }
    head_kernel<<<(BATCH + 255) / 256, 256, 0, stream>>>(h, cl_W2, cl_b2,
                                                         (float*)d_out);
}